// GraphEmbedding_3195455668883
// MI455X (gfx1250) — compile-verified
//
#include <hip/hip_runtime.h>

// SAGEConv (mean aggregation) for MI455X / gfx1250.
// N=100000, E=1600000, IN_C=OUT_C=128.
//
// Phase 1: zero summed (reuses d_out, 51.2MB) + deg (d_ws, 400KB)
// Phase 2: per-edge scatter-add, one wave32 per edge, float4 per lane
// Phase 3: fused GEMM  out = relu([agg | x] @ [W_l ; W_r] + b_l)
//          using V_WMMA_F32_16X16X4_F32 (full-precision f32 WMMA).

#define CIN   128
#define KTOT  256   // concatenated K = 2*CIN
#define APAD  4     // LDS pad: row stride 260 floats -> bank stride 4, conflict-free

typedef __attribute__((ext_vector_type(2))) float v2f;
typedef __attribute__((ext_vector_type(8))) float v8f;

__global__ void zero_kernel(float* __restrict__ summed, float* __restrict__ deg,
                            int n_sum, int n_deg) {
    int i = blockIdx.x * blockDim.x + threadIdx.x;
    int stride = gridDim.x * blockDim.x;
    for (int k = i; k < n_sum; k += stride) summed[k] = 0.0f;
    for (int k = i; k < n_deg; k += stride) deg[k] = 0.0f;
}

// One wave (32 lanes) per edge: lane handles 4 channels (4*32 = 128).
__global__ void scatter_kernel(const float* __restrict__ x,
                               const int* __restrict__ erow,
                               const int* __restrict__ ecol,
                               float* __restrict__ summed,
                               float* __restrict__ deg, int E) {
    int gtid = blockIdx.x * blockDim.x + threadIdx.x;
    int e    = gtid >> 5;
    int lane = gtid & 31;
    if (e >= E) return;
    int dst = erow[e];
    int src = ecol[e];
    const float4* xs = (const float4*)(x + (size_t)src * CIN);
    float4 v = xs[lane];                       // coalesced 16B/lane
    float* sd = summed + (size_t)dst * CIN + lane * 4;
    atomicAdd(sd + 0, v.x);
    atomicAdd(sd + 1, v.y);
    atomicAdd(sd + 2, v.z);
    atomicAdd(sd + 3, v.w);
    if (lane == 0) atomicAdd(deg + dst, 1.0f);
}

// 256 threads = 8 wave32s. Block owns 16 node rows; wave w owns out cols [16w,16w+16).
// A (16x256 f32) staged in LDS; B read from global (W fits in L2).
__global__ void __launch_bounds__(256) gemm_kernel(
    const float* __restrict__ x,
    const float* __restrict__ Wl,
    const float* __restrict__ bl,
    const float* __restrict__ Wr,
    const float* __restrict__ deg,
    float* __restrict__ out /* holds summed on entry for this block's rows */) {

    __shared__ float As[16][KTOT + APAD];
    __shared__ float inv[16];

    const int tid    = threadIdx.x;
    const int m_base = blockIdx.x * 16;

    if (tid < 16) {
        float d = deg[m_base + tid];
        inv[tid] = 1.0f / fmaxf(d, 1.0f);
    }
    __syncthreads();

    // Stage A = [summed * inv_deg | x]  (out currently holds summed for these rows)
    for (int idx = tid; idx < 16 * KTOT; idx += 256) {
        int r = idx >> 8;          // 0..15   (KTOT == 256)
        int k = idx & 255;         // 0..255
        int row = m_base + r;
        float v;
        if (k < CIN) v = out[(size_t)row * CIN + k] * inv[r];
        else         v = x[(size_t)row * CIN + (k - CIN)];
        As[r][k] = v;
    }
    __syncthreads();

    const int wave = tid >> 5;
    const int lane = tid & 31;
    const int half = lane >> 4;    // 0: lanes 0-15, 1: lanes 16-31
    const int mrow = lane & 15;    // A row this lane feeds
    const int n    = wave * 16 + (lane & 15);   // output column

    __builtin_prefetch(Wl + n, 0, 0);
    __builtin_prefetch(Wr + n, 0, 0);

    // A 16x4 f32 layout: VGPR0 = K{0|2}, VGPR1 = K{1|3} by lane-half.
    // B 4x16 f32 layout mirrors it: b.x = row K{0|2}, b.y = row K{1|3}.
    v8f acc = {0.f, 0.f, 0.f, 0.f, 0.f, 0.f, 0.f, 0.f};

    #pragma unroll 4
    for (int t = 0; t < 32; ++t) {             // K = 0..127 : agg @ W_l
        int ks = t * 4 + 2 * half;
        v2f a; a.x = As[mrow][ks];       a.y = As[mrow][ks + 1];
        v2f b; b.x = Wl[ks * CIN + n];   b.y = Wl[(ks + 1) * CIN + n];
        acc = __builtin_amdgcn_wmma_f32_16x16x4_f32(
            false, a, false, b, (short)0, acc, false, false);
    }
    #pragma unroll 4
    for (int t = 0; t < 32; ++t) {             // K = 128..255 : x @ W_r
        int ks = t * 4 + 2 * half;
        v2f a; a.x = As[mrow][CIN + ks]; a.y = As[mrow][CIN + ks + 1];
        v2f b; b.x = Wr[ks * CIN + n];   b.y = Wr[(ks + 1) * CIN + n];
        acc = __builtin_amdgcn_wmma_f32_16x16x4_f32(
            false, a, false, b, (short)0, acc, false, false);
    }

    // D layout: VGPR r holds M = r + 8*half, N = lane&15. Fuse bias + ReLU.
    float bias = bl[n];
    #pragma unroll
    for (int r = 0; r < 8; ++r) {
        int m = m_base + r + 8 * half;
        float v = acc[r] + bias;
        out[(size_t)m * CIN + n] = fmaxf(v, 0.0f);
    }
}

extern "C" void kernel_launch(void* const* d_in, const int* in_sizes, int n_in,
                              void* d_out, int out_size, void* d_ws, size_t ws_size,
                              hipStream_t stream) {
    const float* x    = (const float*)d_in[0];
    const int*   erow = (const int*)d_in[1];
    const int*   ecol = (const int*)d_in[2];
    const float* Wl   = (const float*)d_in[3];
    const float* bl   = (const float*)d_in[4];
    const float* Wr   = (const float*)d_in[5];

    float* out = (float*)d_out;          // doubles as `summed` accumulator
    float* deg = (float*)d_ws;           // N floats

    const int N = in_sizes[0] / CIN;     // 100000
    const int E = in_sizes[1];           // 1600000

    zero_kernel<<<2048, 256, 0, stream>>>(out, deg, N * CIN, N);

    long long threads = (long long)E * 32;
    int sblocks = (int)((threads + 255) / 256);
    scatter_kernel<<<sblocks, 256, 0, stream>>>(x, erow, ecol, out, deg, E);

    gemm_kernel<<<N / 16, 256, 0, stream>>>(x, Wl, bl, Wr, deg, out);
}